// QuanvolutionNet_65481071401520
// MI455X (gfx1250) — compile-verified
//
#include <hip/hip_runtime.h>
#include <hip/hip_bf16.h>

typedef __attribute__((ext_vector_type(2))) float v2f;
typedef __attribute__((ext_vector_type(8))) float v8f;

#define NB   1024
#define NP   196
#define NCLS 10
#define FEAT 784   // NP*4

// workspace layout (in floats):
//  [0, 1024)          : M   (4 matrices x 16 x 16)
//  [1024, 803840)     : feats (1024 x 784)
//  [803840, 820224)   : logits padded (1024 x 16)
#define WS_M      0
#define WS_FEATS  1024
#define WS_LOGITS 803840

// ---------------------------------------------------------------------------
// Kernel 1: build the fixed 16x16 unitary V from var_params, then the four
// real symmetric matrices M_w[j,j'] = sum_k s_w(k) * Re(V[k,j] * conj(V[k,j'])).
// Wire w's bit in the 16-state index is bit (3-w).
// ---------------------------------------------------------------------------
__global__ __launch_bounds__(256) void k_build_M(const float* __restrict__ vp,
                                                 float* __restrict__ ws) {
  __shared__ float Vr[16][16];
  __shared__ float Vi[16][16];
  const int tid = threadIdx.x;
  if (tid < 16) {
    // thread `tid` evolves column tid of V (starts as e_tid)
    float vr[16], vi[16];
#pragma unroll
    for (int i = 0; i < 16; ++i) { vr[i] = (i == tid) ? 1.f : 0.f; vi[i] = 0.f; }
#pragma unroll
    for (int l = 0; l < 2; ++l) {
#pragma unroll
      for (int w = 0; w < 4; ++w) {
        const int m = 1 << (3 - w);
        const float tx = vp[(l*4 + w)*3 + 0];
        const float tz = vp[(l*4 + w)*3 + 1];
        const float ty = vp[(l*4 + w)*3 + 2];
        const float cx = cosf(0.5f*tx), sx = sinf(0.5f*tx);
        const float cz = cosf(0.5f*tz), sz = sinf(0.5f*tz);
        const float cy = cosf(0.5f*ty), sy = sinf(0.5f*ty);
#pragma unroll
        for (int i0 = 0; i0 < 16; ++i0) {
          if (i0 & m) continue;
          const int i1 = i0 | m;
          float r0 = vr[i0], im0 = vi[i0], r1 = vr[i1], im1 = vi[i1];
          // RX: [[c,-is],[-is,c]]
          float ar = cx*r0 + sx*im1,  ai = cx*im0 - sx*r1;
          float br = sx*im0 + cx*r1,  bi = -sx*r0 + cx*im1;
          // RZ: diag(e^{-it/2}, e^{+it/2})
          r0 = cz*ar + sz*ai;  im0 = cz*ai - sz*ar;
          r1 = cz*br - sz*bi;  im1 = cz*bi + sz*br;
          // RY: [[c,-s],[s,c]] (real)
          vr[i0] = cy*r0 - sy*r1;  vi[i0] = cy*im0 - sy*im1;
          vr[i1] = sy*r0 + cy*r1;  vi[i1] = sy*im0 + cy*im1;
        }
      }
      // CNOT chain (0,1),(1,2),(2,3),(3,0); wire w -> bit mask 1<<(3-w)
      const int cm[4] = {8, 4, 2, 1};
      const int tm[4] = {4, 2, 1, 8};
#pragma unroll
      for (int g = 0; g < 4; ++g) {
#pragma unroll
        for (int i = 0; i < 16; ++i) {
          if ((i & cm[g]) && !(i & tm[g])) {
            const int j2 = i | tm[g];
            float t;
            t = vr[i]; vr[i] = vr[j2]; vr[j2] = t;
            t = vi[i]; vi[i] = vi[j2]; vi[j2] = t;
          }
        }
      }
    }
#pragma unroll
    for (int i = 0; i < 16; ++i) { Vr[i][tid] = vr[i]; Vi[i][tid] = vi[i]; }
  }
  __syncthreads();
  const int j = tid >> 4, jp = tid & 15;
  float a0 = 0.f, a1 = 0.f, a2 = 0.f, a3 = 0.f;
#pragma unroll
  for (int k = 0; k < 16; ++k) {
    const float re = Vr[k][j]*Vr[k][jp] + Vi[k][j]*Vi[k][jp];
    a0 += ((k >> 3) & 1) ? -re : re;
    a1 += ((k >> 2) & 1) ? -re : re;
    a2 += ((k >> 1) & 1) ? -re : re;
    a3 += ( k       & 1) ? -re : re;
  }
  ws[WS_M + 0*256 + j*16 + jp] = a0;
  ws[WS_M + 1*256 + j*16 + jp] = a1;
  ws[WS_M + 2*256 + j*16 + jp] = a2;
  ws[WS_M + 3*256 + j*16 + jp] = a3;
}

// ---------------------------------------------------------------------------
// Kernel 2: one wave handles 16 (b,p) pairs. Build PHI (16 states x 16 pairs)
// directly in the WMMA B layout (lane-half K split realized with arithmetic
// blends, NOT selects, to avoid dynamic-index cndmask chains), chain
// 4x V_WMMA_F32_16X16X4_F32 to get T_w = M_w * PHI, then
// z_w[n] = sum_j PHI[j,n]*T_w[j,n] via per-lane row-reduce + one xor-16
// shuffle. feats[pair*4+w] = scale * z_w.
// ---------------------------------------------------------------------------
__global__ __launch_bounds__(256) void k_quantum(const float* __restrict__ x,
                                                 const float* __restrict__ scale,
                                                 const float* __restrict__ Mbase,
                                                 float* __restrict__ feats) {
  const int lane = threadIdx.x & 31;
  const int wave = threadIdx.x >> 5;
  const unsigned tile = blockIdx.x * 8u + (unsigned)wave;   // 12544 tiles total
  const unsigned n    = (unsigned)(lane & 15);              // column (pair-in-tile)
  const unsigned pair = tile * 16u + n;
  const unsigned b  = pair / 196u;
  const unsigned p  = pair % 196u;
  const unsigned pr = p / 14u, pc = p % 14u;
  const float* xb = x + b * 784u;

  float cw[4], sw[4];
#pragma unroll
  for (int w = 0; w < 4; ++w) {
    const float t = xb[(2u*pr + (unsigned)(w >> 1))*28u + (2u*pc + (unsigned)(w & 1))];
    cw[w] = cosf(0.5f * t);
    sw[w] = sinf(0.5f * t);
  }
  float phi[16];
#pragma unroll
  for (int j = 0; j < 16; ++j) {
    phi[j] = (((j >> 3) & 1) ? sw[0] : cw[0]) * (((j >> 2) & 1) ? sw[1] : cw[1])
           * (((j >> 1) & 1) ? sw[2] : cw[2]) * ((j & 1)        ? sw[3] : cw[3]);
  }
  const float scl = scale[0];
  const float hi  = (lane >= 16) ? 1.f : 0.f;   // lane-half as a float
  const int   off0 = (lane >= 16) ? 2 : 0;      // only used in addresses

  // B-operand values (same for all 4 wires): step s needs phi[4s+2h+{0,1}]
  float bx[4], by[4];
#pragma unroll
  for (int s = 0; s < 4; ++s) {
    bx[s] = fmaf(hi, phi[4*s + 2] - phi[4*s + 0], phi[4*s + 0]);
    by[s] = fmaf(hi, phi[4*s + 3] - phi[4*s + 1], phi[4*s + 1]);
  }
  // reduction weights: lanes 0-15 cover j=r, lanes 16-31 cover j=8+r
  float pw[8];
#pragma unroll
  for (int r = 0; r < 8; ++r)
    pw[r] = fmaf(hi, phi[8 + r] - phi[r], phi[r]);

#pragma unroll
  for (int w = 0; w < 4; ++w) {
    const float* Mrow = Mbase + w*256 + (lane & 15)*16 + off0;
    v8f acc = {};
#pragma unroll
    for (int s = 0; s < 4; ++s) {
      const v2f a = *reinterpret_cast<const v2f*>(Mrow + 4*s);  // 8B aligned
      v2f bb; bb.x = bx[s]; bb.y = by[s];
      acc = __builtin_amdgcn_wmma_f32_16x16x4_f32(false, a, false, bb,
                                                  (short)0, acc, false, false);
    }
    float part = 0.f;
#pragma unroll
    for (int r = 0; r < 8; ++r)
      part = fmaf(acc[r], pw[r], part);
    part += __shfl_xor(part, 16, 32);
    if (lane < 16) feats[pair * 4u + (unsigned)w] = scl * part;
  }
}

// ---------------------------------------------------------------------------
// Kernel 3: logits = feats(1024x784) @ fc_w^T(784x10) + b, classes padded to
// 16 via zero-mask multiply (keeps EXEC all-ones around the WMMAs).
// One wave = 16 batch rows; K chained over 784 in steps of 4.
// ---------------------------------------------------------------------------
__global__ __launch_bounds__(256) void k_fc(const float* __restrict__ feats,
                                            const float* __restrict__ fc_w,
                                            const float* __restrict__ fc_b,
                                            float* __restrict__ logits) {
  const int lane  = threadIdx.x & 31;
  const int wave  = threadIdx.x >> 5;
  const int mtile = blockIdx.x * 8 + wave;     // 0..63
  const int  n    = lane & 15;                 // class column (padded)
  const int  row  = mtile * 16 + (lane & 15);  // batch row for A operand
  const int  off0 = (lane >= 16) ? 2 : 0;
  const float* Arow = feats + row * 784 + off0;
  const float* Bcol = fc_w + (n < 10 ? n : 9) * 784 + off0;  // clamp: always valid
  const float  msk  = (n < 10) ? 1.f : 0.f;

  v8f acc = {};
#pragma unroll 4
  for (int kb = 0; kb < 784; kb += 4) {
    const v2f a  = *reinterpret_cast<const v2f*>(Arow + kb);   // 8B aligned
    const v2f bv = *reinterpret_cast<const v2f*>(Bcol + kb);
    v2f bb; bb.x = msk * bv.x; bb.y = msk * bv.y;
    acc = __builtin_amdgcn_wmma_f32_16x16x4_f32(false, a, false, bb,
                                                (short)0, acc, false, false);
  }
  const float bias = (n < 10) ? fc_b[n] : 0.f;
  const int   mo   = (lane >= 16) ? 8 : 0;
#pragma unroll
  for (int r = 0; r < 8; ++r)
    logits[(mtile*16 + mo + r) * 16 + n] = acc[r] + bias;
}

// ---------------------------------------------------------------------------
// Kernel 4: row-wise log_softmax over the 10 valid classes.
// ---------------------------------------------------------------------------
__global__ __launch_bounds__(256) void k_lsm(const float* __restrict__ logits,
                                             float* __restrict__ out) {
  const int b = blockIdx.x * 256 + threadIdx.x;
  if (b >= NB) return;
  const float* lp = logits + b * 16;
  float mx = lp[0];
#pragma unroll
  for (int c = 1; c < NCLS; ++c) mx = fmaxf(mx, lp[c]);
  float s = 0.f;
#pragma unroll
  for (int c = 0; c < NCLS; ++c) s += expf(lp[c] - mx);
  const float lse = mx + logf(s);
#pragma unroll
  for (int c = 0; c < NCLS; ++c) out[b * NCLS + c] = lp[c] - lse;
}

extern "C" void kernel_launch(void* const* d_in, const int* in_sizes, int n_in,
                              void* d_out, int out_size, void* d_ws, size_t ws_size,
                              hipStream_t stream) {
  (void)in_sizes; (void)n_in; (void)out_size; (void)ws_size;
  const float* x     = (const float*)d_in[0];  // (1024,1,28,28)
  const float* vp    = (const float*)d_in[1];  // (2,4,3)
  const float* scale = (const float*)d_in[2];  // (1,)
  const float* fc_w  = (const float*)d_in[3];  // (10,784)
  const float* fc_b  = (const float*)d_in[4];  // (10,)
  float* ws  = (float*)d_ws;
  float* out = (float*)d_out;

  hipLaunchKernelGGL(k_build_M, dim3(1),    dim3(256), 0, stream, vp, ws);
  hipLaunchKernelGGL(k_quantum, dim3(1568), dim3(256), 0, stream,
                     x, scale, ws + WS_M, ws + WS_FEATS);
  hipLaunchKernelGGL(k_fc,      dim3(8),    dim3(256), 0, stream,
                     ws + WS_FEATS, fc_w, fc_b, ws + WS_LOGITS);
  hipLaunchKernelGGL(k_lsm,     dim3(4),    dim3(256), 0, stream,
                     ws + WS_LOGITS, out);
}